// GNNAutoencoder_34686155882552
// MI455X (gfx1250) — compile-verified
//
#include <hip/hip_runtime.h>
#include <hip/hip_bf16.h>

#define N_NODES 50000
#define N_EDGES 800000
#define N_GRAPHS 256
#define HID 64
#define EMB 3
#define OUT_DIM 7
#define NEG_SLOPE 0.01f

typedef float v2f __attribute__((ext_vector_type(2)));
typedef float v8f __attribute__((ext_vector_type(8)));

__device__ __forceinline__ float lrelu(float v) {
    return (v >= 0.f) ? v : NEG_SLOPE * v;
}

// ---------------- embedding gather: h0[i,c] = emb[x[i], c] ----------------
__global__ void k_embed(const int* __restrict__ x, const float* __restrict__ emb,
                        float* __restrict__ h0) {
    int idx = blockIdx.x * blockDim.x + threadIdx.x;
    if (idx >= N_NODES * EMB) return;
    int i = idx / EMB, c = idx % EMB;
    h0[idx] = emb[x[i] * EMB + c];
}

// ---------------- edge scatter-add: aggr[dst] += X[src] ----------------
template <int VEC>
__global__ void k_scatter_add(const float* __restrict__ X, const int* __restrict__ src,
                              const int* __restrict__ dst, float* __restrict__ aggr,
                              int fin, long long total_chunks) {
    long long idx = (long long)blockIdx.x * blockDim.x + threadIdx.x;
    if (idx >= total_chunks) return;
    int cpe = fin / VEC;
    long long e = idx / cpe;
    int c = (int)(idx % cpe) * VEC;
    int s = src[e], d = dst[e];
    const float* xp = X + (long long)s * fin + c;
    float* ap = aggr + (long long)d * fin + c;
    if (VEC == 4) {
        float4 v = *(const float4*)xp;
        atomicAdd(ap + 0, v.x);
        atomicAdd(ap + 1, v.y);
        atomicAdd(ap + 2, v.z);
        atomicAdd(ap + 3, v.w);
    } else {
        atomicAdd(ap, *xp);
    }
}

// ---------------- conv1: tiny K=3, plain VALU ----------------
__global__ void k_conv1(const float* __restrict__ aggr3, const float* __restrict__ h0,
                        const float* __restrict__ w_rel, const float* __restrict__ w_root,
                        const float* __restrict__ b, float* __restrict__ h1) {
    int idx = blockIdx.x * blockDim.x + threadIdx.x;
    if (idx >= N_NODES * HID) return;
    int i = idx / HID, o = idx % HID;
    float s = b[o];
#pragma unroll
    for (int k = 0; k < EMB; ++k) {
        s += aggr3[i * EMB + k] * w_rel[k * HID + o];
        s += h0[i * EMB + k] * w_root[k * HID + o];
    }
    h1[idx] = lrelu(s);
}

// -------- weight repack into WMMA B-fragment order --------
// Wp[(q*Fout + n)*4 + j] = W[(4q + j)*Fout + n]
__global__ void k_pack_w(const float* __restrict__ W, float* __restrict__ Wp,
                         int Fin, int Fout) {
    long long idx = (long long)blockIdx.x * blockDim.x + threadIdx.x;
    if (idx >= (long long)Fin * Fout) return;
    int kk = (int)(idx / Fout);
    int n = (int)(idx - (long long)kk * Fout);
    Wp[((long long)(kk >> 2) * Fout + n) * 4 + (kk & 3)] = W[idx];
}

// ------- fused dual GEMM + bias + leaky relu (f32 WMMA, LDS-staged B) -------
// out[M,Fout] = lrelu(A1 @ W1 + A2 @ W2 + bias), W1p/W2p pre-packed.
// Block = 8 waves sharing one 64-wide N strip (tg); wave w owns M tile bm*8+w.
// B is staged per 64-K chunk in LDS (2 x 16 KB), loaded once per block.
// Inner K-step: 8 ds_load_b64 (B) + prefetched global A + 8 WMMAs, dependent
// WMMAs on one accumulator distance-4 apart.
// A fragment (16x4 f32): lanes 0-15 K=k0,k0+1 of row M=lane; lanes 16-31 K=k0+2,k0+3.
// C/D: vgpr r -> (M = r + 8*(lane>=16), N = lane&15).
__global__ void __launch_bounds__(256) k_dual_gemm_lrelu(
    const float* __restrict__ A1, const float* __restrict__ A2,
    const float* __restrict__ W1p, const float* __restrict__ W2p,
    const float* __restrict__ bias, float* __restrict__ out,
    int M, int Fin, int Fout) {
    __shared__ __align__(16) float lds_b1[16 * 256];
    __shared__ __align__(16) float lds_b2[16 * 256];

    const int tid = threadIdx.x;
    const int lane = tid & 31;
    const int wave = tid >> 5;
    const int ngroups = Fout >> 6;  // 64-wide N strips
    const int bm = blockIdx.x / ngroups;
    const int tg = blockIdx.x - bm * ngroups;
    const int tm = bm * 8 + wave;
    const bool valid = tm < (M >> 4);  // uniform per wave

    const int n0 = tg << 6;
    const int row = lane & 15;
    const int khalf = (lane >> 4) << 1;  // 0 for lanes 0-15, 2 for lanes 16-31
    const int col = lane & 15;
    const int lb = col * 4 + khalf;  // per-lane LDS fragment offset (elements)
    const long long a_row = valid ? (long long)((tm << 4) + row) * Fin : 0;
    const float* a1base = A1 + a_row + khalf;
    const float* a2base = A2 + a_row + khalf;

    v8f acc[4];
#pragma unroll
    for (int t = 0; t < 4; ++t) acc[t] = (v8f){0.f, 0.f, 0.f, 0.f, 0.f, 0.f, 0.f, 0.f};

    const int nchunks = Fin >> 6;  // 64 K-values (16 quads) per chunk
    for (int ch = 0; ch < nchunks; ++ch) {
        __syncthreads();  // previous chunk's compute done before overwrite
        // cooperative fill: 16 quads x 64 cols x 4 packed = 4096 floats per matrix
        const long long wq0 = ((long long)(ch << 4) * Fout + n0) * 4;
        for (int i = tid; i < 1024; i += 256) {
            const int q = i >> 6, f4 = i & 63;
            const long long srcoff = wq0 + (long long)q * Fout * 4 + f4 * 4;
            *(float4*)&lds_b1[q * 256 + f4 * 4] = *(const float4*)(W1p + srcoff);
            *(float4*)&lds_b2[q * 256 + f4 * 4] = *(const float4*)(W2p + srcoff);
        }
        __syncthreads();

        if (valid) {
            const float* ap1 = a1base + (ch << 6);
            const float* ap2 = a2base + (ch << 6);
            v2f a1 = *(const v2f*)ap1;
            v2f a2 = *(const v2f*)ap2;
#pragma unroll 4
            for (int q = 0; q < 16; ++q) {
                v2f na1 = a1, na2 = a2;
                if (q < 15) {  // prefetch next K-step's A (stays within chunk)
                    na1 = *(const v2f*)(ap1 + ((q + 1) << 2));
                    na2 = *(const v2f*)(ap2 + ((q + 1) << 2));
                }
                v2f b1[4], b2[4];
#pragma unroll
                for (int t = 0; t < 4; ++t) {
                    b1[t] = *(const v2f*)&lds_b1[q * 256 + t * 64 + lb];
                    b2[t] = *(const v2f*)&lds_b2[q * 256 + t * 64 + lb];
                }
#pragma unroll
                for (int t = 0; t < 4; ++t)
                    acc[t] = __builtin_amdgcn_wmma_f32_16x16x4_f32(
                        false, a1, false, b1[t], (short)0, acc[t], false, false);
#pragma unroll
                for (int t = 0; t < 4; ++t)
                    acc[t] = __builtin_amdgcn_wmma_f32_16x16x4_f32(
                        false, a2, false, b2[t], (short)0, acc[t], false, false);
                a1 = na1;
                a2 = na2;
            }
        }
    }

    if (valid) {
        const int moff = (lane >> 4) << 3;  // 0 or 8
        const long long out_row0 = (long long)((tm << 4) + moff) * Fout;
#pragma unroll
        for (int t = 0; t < 4; ++t) {
            const int nc = n0 + t * 16 + col;
            const float bv = bias[nc];
#pragma unroll
            for (int r = 0; r < 8; ++r) {
                float v = acc[t][r] + bv;
                out[out_row0 + (long long)r * Fout + nc] = lrelu(v);
            }
        }
    }
}

// ---------------- pooling ----------------
__global__ void k_counts(const int* __restrict__ batch, float* __restrict__ counts) {
    int i = blockIdx.x * blockDim.x + threadIdx.x;
    if (i >= N_NODES) return;
    atomicAdd(&counts[batch[i]], 1.0f);
}

__global__ void k_pool_accum(const float* __restrict__ latent, const int* __restrict__ batch,
                             float* __restrict__ pooled) {
    int idx = blockIdx.x * blockDim.x + threadIdx.x;
    if (idx >= N_NODES * 256) return;
    int i = idx >> 8, c = idx & 255;
    atomicAdd(&pooled[batch[i] * 256 + c], latent[idx]);
}

__global__ void k_logits(const float* __restrict__ pooled, const float* __restrict__ counts,
                         const float* __restrict__ w_lin, const float* __restrict__ b_lin,
                         float* __restrict__ logits) {
    int idx = blockIdx.x * blockDim.x + threadIdx.x;
    if (idx >= N_GRAPHS * OUT_DIM) return;
    int g = idx / OUT_DIM, o = idx % OUT_DIM;
    float inv = 1.0f / fmaxf(counts[g], 1.0f);
    float s = b_lin[o];
    for (int c = 0; c < 256; ++c) s += pooled[g * 256 + c] * inv * w_lin[c * OUT_DIM + o];
    logits[idx] = s;
}

static inline int cdiv(long long a, int b) { return (int)((a + b - 1) / b); }

extern "C" void kernel_launch(void* const* d_in, const int* in_sizes, int n_in,
                              void* d_out, int out_size, void* d_ws, size_t ws_size,
                              hipStream_t stream) {
    // inputs (setup_inputs order)
    const int*   x       = (const int*)d_in[0];
    const int*   eidx    = (const int*)d_in[1];
    const int*   batch   = (const int*)d_in[2];
    const float* emb     = (const float*)d_in[3];
    const float* w1_rel  = (const float*)d_in[4];
    const float* w1_root = (const float*)d_in[5];
    const float* b1      = (const float*)d_in[6];
    const float* w2_rel  = (const float*)d_in[7];
    const float* w2_root = (const float*)d_in[8];
    const float* b2      = (const float*)d_in[9];
    const float* we_rel  = (const float*)d_in[10];
    const float* we_root = (const float*)d_in[11];
    const float* be      = (const float*)d_in[12];
    const float* wd_rel  = (const float*)d_in[13];
    const float* wd_root = (const float*)d_in[14];
    const float* bd      = (const float*)d_in[15];
    const float* w_lin   = (const float*)d_in[16];
    const float* b_lin   = (const float*)d_in[17];

    const int* src = eidx;             // edge_index[0]
    const int* dst = eidx + N_EDGES;   // edge_index[1]

    // outputs: (logits[256*7], reconstructed[N*1024], original[N*1024])
    float* out_f  = (float*)d_out;
    float* logits = out_f;
    float* recon  = out_f + N_GRAPHS * OUT_DIM;
    float* orig   = recon + (long long)N_NODES * 1024;

    // workspace layout (256B-aligned element offsets)
    float* ws = (float*)d_ws;
    long long off = 0;
    float* h0 = ws + off;      off += ((long long)N_NODES * EMB + 63) / 64 * 64;
    float* h1 = ws + off;      off += (long long)N_NODES * HID;
    float* aggr = ws + off;    off += (long long)N_NODES * 1024;  // reused each layer
    float* latent = ws + off;  off += (long long)N_NODES * 256;
    float* pooled = ws + off;  off += N_GRAPHS * 256;
    float* counts = ws + off;  off += 64;
    float* w2p_rel  = ws + off; off += (long long)HID * 1024;
    float* w2p_root = ws + off; off += (long long)HID * 1024;
    float* wep_rel  = ws + off; off += (long long)1024 * 256;
    float* wep_root = ws + off; off += (long long)1024 * 256;
    float* wdp_rel  = ws + off; off += (long long)256 * 1024;
    float* wdp_root = ws + off; off += (long long)256 * 1024;

    const int T = 256;
    const int MT = N_NODES / 16;      // 3125 M tiles
    const int MB = (MT + 7) / 8;      // blocks along M (8 waves each)

    // ---- pack weights into B-fragment order (tiny, one pass) ----
    k_pack_w<<<cdiv((long long)HID * 1024, T), T, 0, stream>>>(w2_rel, w2p_rel, HID, 1024);
    k_pack_w<<<cdiv((long long)HID * 1024, T), T, 0, stream>>>(w2_root, w2p_root, HID, 1024);
    k_pack_w<<<cdiv((long long)1024 * 256, T), T, 0, stream>>>(we_rel, wep_rel, 1024, 256);
    k_pack_w<<<cdiv((long long)1024 * 256, T), T, 0, stream>>>(we_root, wep_root, 1024, 256);
    k_pack_w<<<cdiv((long long)256 * 1024, T), T, 0, stream>>>(wd_rel, wdp_rel, 256, 1024);
    k_pack_w<<<cdiv((long long)256 * 1024, T), T, 0, stream>>>(wd_root, wdp_root, 256, 1024);

    // ---- layer 1: emb -> h1[N,64] ----
    k_embed<<<cdiv((long long)N_NODES * EMB, T), T, 0, stream>>>(x, emb, h0);
    hipMemsetAsync(aggr, 0, (size_t)N_NODES * EMB * sizeof(float), stream);
    {
        long long chunks = (long long)N_EDGES * EMB;
        k_scatter_add<1><<<cdiv(chunks, T), T, 0, stream>>>(h0, src, dst, aggr, EMB, chunks);
    }
    k_conv1<<<cdiv((long long)N_NODES * HID, T), T, 0, stream>>>(aggr, h0, w1_rel, w1_root, b1, h1);

    // ---- layer 2 (conv2): h1[N,64] -> original[N,1024] ----
    hipMemsetAsync(aggr, 0, (size_t)N_NODES * HID * sizeof(float), stream);
    {
        long long chunks = (long long)N_EDGES * (HID / 4);
        k_scatter_add<4><<<cdiv(chunks, T), T, 0, stream>>>(h1, src, dst, aggr, HID, chunks);
    }
    k_dual_gemm_lrelu<<<MB * (1024 / 64), T, 0, stream>>>(aggr, h1, w2p_rel, w2p_root, b2,
                                                          orig, N_NODES, HID, 1024);

    // ---- layer 3 (enc): original[N,1024] -> latent[N,256] ----
    hipMemsetAsync(aggr, 0, (size_t)N_NODES * 1024 * sizeof(float), stream);
    {
        long long chunks = (long long)N_EDGES * (1024 / 4);
        k_scatter_add<4><<<cdiv(chunks, T), T, 0, stream>>>(orig, src, dst, aggr, 1024, chunks);
    }
    k_dual_gemm_lrelu<<<MB * (256 / 64), T, 0, stream>>>(aggr, orig, wep_rel, wep_root, be,
                                                         latent, N_NODES, 1024, 256);

    // ---- pooling + logits ----
    hipMemsetAsync(pooled, 0, (size_t)(N_GRAPHS * 256 + 64) * sizeof(float), stream);
    k_counts<<<cdiv(N_NODES, T), T, 0, stream>>>(batch, counts);
    k_pool_accum<<<cdiv((long long)N_NODES * 256, T), T, 0, stream>>>(latent, batch, pooled);
    k_logits<<<cdiv(N_GRAPHS * OUT_DIM, T), T, 0, stream>>>(pooled, counts, w_lin, b_lin, logits);

    // ---- layer 4 (dec): latent[N,256] -> reconstructed[N,1024] ----
    hipMemsetAsync(aggr, 0, (size_t)N_NODES * 256 * sizeof(float), stream);
    {
        long long chunks = (long long)N_EDGES * (256 / 4);
        k_scatter_add<4><<<cdiv(chunks, T), T, 0, stream>>>(latent, src, dst, aggr, 256, chunks);
    }
    k_dual_gemm_lrelu<<<MB * (1024 / 64), T, 0, stream>>>(aggr, latent, wdp_rel, wdp_root, bd,
                                                          recon, N_NODES, 256, 1024);
}